// Slice_60662118088797
// MI455X (gfx1250) — compile-verified
//
#include <hip/hip_runtime.h>
#include <stdint.h>

// Problem constants (fixed by the reference)
#define BH   16          // B*H = 4*4
#define NN   32768       // N
#define FF   32          // feature_dim = C/H
#define SS   8           // SPREAD
#define W3   32768       // W^3

typedef float    v4f __attribute__((ext_vector_type(4)));
typedef unsigned v4u __attribute__((ext_vector_type(4)));

// ---------------------------------------------------------------------------
// Kernel 1: per-(b,h) transpose  convolved[bh][F][W3] -> convT[bh][W3][F]
// so one gathered index yields a contiguous 128B channel vector.
// b128 on both global phases; input NT (read once), output RT (L2-resident
// for the gather kernel). LDS tile padded to 257 floats.
// ---------------------------------------------------------------------------
__global__ __launch_bounds__(256) void transpose_fm(const float* __restrict__ in,
                                                    float* __restrict__ outT) {
    __shared__ float tile[FF][257];
    const int slab = blockIdx.x;                  // 2048 slabs
    const int bh   = slab >> 7;                   // 128 slabs of 256 cols per bh
    const int m0   = (slab & 127) << 8;           // *256
    const int tid  = threadIdx.x;

    const v4f* ib = (const v4f*)(in + (size_t)bh * FF * W3 + m0);
#pragma unroll
    for (int it = 0; it < 8; ++it) {
        const int q  = it * 256 + tid;            // 2048 float4 per slab
        const int f  = q >> 6;                    // 64 float4 per row
        const int c4 = q & 63;
        v4f v = __builtin_nontemporal_load(&ib[(size_t)f * (W3 / 4) + c4]);
        tile[f][c4 * 4 + 0] = v[0];
        tile[f][c4 * 4 + 1] = v[1];
        tile[f][c4 * 4 + 2] = v[2];
        tile[f][c4 * 4 + 3] = v[3];
    }
    __syncthreads();

    // output slab [256 m][32 f] is contiguous -> linear b128 writes
    v4f* ob = (v4f*)(outT + ((size_t)bh * W3 + m0) * FF);
#pragma unroll
    for (int it = 0; it < 8; ++it) {
        const int q  = it * 256 + tid;            // float4 index
        const int f0 = (q & 7) << 2;
        const int mp = q >> 3;
        v4f v;
        v[0] = tile[f0 + 0][mp];                  // banks (4a+b) distinct: conflict-free
        v[1] = tile[f0 + 1][mp];
        v[2] = tile[f0 + 2][mp];
        v[3] = tile[f0 + 3][mp];
        ob[q] = v;                                // RT: keep convT in L2
    }
}

// ---------------------------------------------------------------------------
// Kernel 2: gather + weighted spread-reduction.
// One wave per 32-n tile; lane = f during accumulation. {idx,weight} pairs
// are brought into LDS with CDNA5 async global->LDS loads (ASYNCcnt) in an
// interleaved per-n layout (20-dword stride: 16B-aligned rows) so the hot
// loop needs only 4 broadcast ds_load_b128 per output column. Each (s,n)
// gather is one fully coalesced 128B load from the L2-resident table.
// Results staged in padded LDS and written out as coalesced b128 NT stores.
// ---------------------------------------------------------------------------
__global__ __launch_bounds__(256) void slice_gather(const float* __restrict__ lc,
                                                    const int*   __restrict__ idx,
                                                    const float* __restrict__ convT,
                                                    float*       __restrict__ out) {
    __shared__ v4u   s_tile[8][32][5];   // [wave][n'][5x16B]; pairs at dword 2s,2s+1
    __shared__ float s_out [8][32][33];  // [wave][n'][f], pad 33 (gcd(33,64)=1)

    const int wave  = threadIdx.x >> 5;
    const int lane  = threadIdx.x & 31;
    const int tile  = blockIdx.x * 8 + wave;   // 16384 tiles
    const int bh    = tile >> 10;              // / (N/32)
    const int nbase = (tile & 1023) << 5;

    const int*   gIdx = idx + (size_t)(bh * SS) * NN + nbase;
    const float* gW   = lc  + (size_t)(bh * SS) * NN + nbase;

    // --- async global -> LDS tile fill (tracked by ASYNCcnt) ---
    const unsigned ldsBase = (unsigned)(uintptr_t)&s_tile[wave][0][0];
#pragma unroll
    for (int s = 0; s < SS; ++s) {
        const unsigned voff = (unsigned)((s * NN + lane) * 4);
        const unsigned dloc = ldsBase + (unsigned)((lane * 20 + 2 * s) * 4);
        asm volatile("global_load_async_to_lds_b32 %0, %1, %2"
                     :: "v"(dloc), "v"(voff), "s"(gIdx) : "memory");
        asm volatile("global_load_async_to_lds_b32 %0, %1, %2"
                     :: "v"(dloc + 4u), "v"(voff), "s"(gW) : "memory");
    }
    asm volatile("s_wait_asynccnt 0" ::: "memory");

    // --- accumulate: per column, 4 broadcast b128 LDS reads + 8 gathers ---
    const float* cbase = convT + (size_t)bh * W3 * FF + lane;
#pragma unroll 4
    for (int nn = 0; nn < 32; ++nn) {
        float acc = 0.0f;
#pragma unroll
        for (int q = 0; q < 4; ++q) {              // s = 2q, 2q+1
            const v4u p = s_tile[wave][nn][q];     // broadcast ds_load_b128
            acc = fmaf(__uint_as_float(p[1]), cbase[(int)p[0] * FF], acc);
            acc = fmaf(__uint_as_float(p[3]), cbase[(int)p[2] * FF], acc);
        }
        s_out[wave][nn][lane] = acc;
    }
    __syncthreads();

    // --- transposed write-out: b128 NT stores, conflict-free LDS reads ---
    float* obase = out + (size_t)(bh * FF) * NN + nbase;
    const int l8 = lane & 7;                       // n-quad selector
    const int lr = lane >> 3;                      // f-row within quad
#pragma unroll
    for (int it = 0; it < 8; ++it) {
        const int f = it * 4 + lr;
        v4f v;
        v[0] = s_out[wave][l8 * 4 + 0][f];
        v[1] = s_out[wave][l8 * 4 + 1][f];
        v[2] = s_out[wave][l8 * 4 + 2][f];
        v[3] = s_out[wave][l8 * 4 + 3][f];
        __builtin_nontemporal_store(v, (v4f*)(obase + (size_t)f * NN + l8 * 4));
    }
}

// ---------------------------------------------------------------------------
// Fallback (workspace too small): direct per-f scalar gathers from the
// original layout. Correct but gather-transaction heavy.
// ---------------------------------------------------------------------------
__global__ __launch_bounds__(256) void slice_direct(const float* __restrict__ lc,
                                                    const int*   __restrict__ idx,
                                                    const float* __restrict__ conv,
                                                    float*       __restrict__ out) {
    const int gid = blockIdx.x * 256 + threadIdx.x;   // over BH*N = 524288
    const int bh  = gid >> 15;
    const int n   = gid & (NN - 1);

    int   id[SS];
    float w [SS];
#pragma unroll
    for (int s = 0; s < SS; ++s) {
        id[s] = idx[(size_t)(bh * SS + s) * NN + n];
        w [s] = lc [(size_t)(bh * SS + s) * NN + n];
    }
    const float* cb = conv + (size_t)bh * FF * W3;
#pragma unroll 4
    for (int f = 0; f < FF; ++f) {
        float acc = 0.0f;
#pragma unroll
        for (int s = 0; s < SS; ++s)
            acc = fmaf(w[s], cb[(size_t)f * W3 + id[s]], acc);
        __builtin_nontemporal_store(acc, out + (size_t)(bh * FF + f) * NN + n);
    }
}

extern "C" void kernel_launch(void* const* d_in, const int* in_sizes, int n_in,
                              void* d_out, int out_size, void* d_ws, size_t ws_size,
                              hipStream_t stream) {
    (void)in_sizes; (void)n_in; (void)out_size;
    const float* lc   = (const float*)d_in[0];
    const int*   fidx = (const int*)d_in[1];
    const float* conv = (const float*)d_in[2];
    float*       out  = (float*)d_out;

    const size_t need = (size_t)BH * W3 * FF * sizeof(float);   // 64 MB
    if (ws_size >= need) {
        float* convT = (float*)d_ws;
        transpose_fm<<<2048, 256, 0, stream>>>(conv, convT);
        slice_gather<<<2048, 256, 0, stream>>>(lc, fidx, convT, out);
    } else {
        slice_direct<<<2048, 256, 0, stream>>>(lc, fidx, conv, out);
    }
}